// Network_82197084111190
// MI455X (gfx1250) — compile-verified
//
#include <hip/hip_runtime.h>

// ---------------------------------------------------------------------------
// GCN (5 layers, D=128) + global max/mean pool + linear head for MI455X.
// GEMMs use V_WMMA_F32_16X16X4_F32 (fp32 WMMA) to keep f32 semantics.
// Edge gather/scatter is the bandwidth-bound phase: float4/lane coalesced
// gathers + L2-resident f32 atomics.
// ---------------------------------------------------------------------------

typedef float v2f __attribute__((ext_vector_type(2)));
typedef float v8f __attribute__((ext_vector_type(8)));

#define DMODEL 128
#define NCLASS 16
#define NGRAPH 512
#define ALPHA  0.01f

__device__ __forceinline__ float leaky(float v) { return v > 0.f ? v : ALPHA * v; }

// order-preserving float->uint encoding for atomicMax-based segment_max
__device__ __forceinline__ unsigned fenc(float x) {
    unsigned b = __float_as_uint(x);
    return (b & 0x80000000u) ? ~b : (b | 0x80000000u);
}
__device__ __forceinline__ float fdec(unsigned k) {
    return __uint_as_float((k & 0x80000000u) ? (k & 0x7FFFFFFFu) : ~k);
}

// ---------------------------------------------------------------------------
// WMMA GEMM: out[rb*16 .. rb*16+15][:] = act(A_rows) @ W + bias
// W is [128][128] row-major, bias [128]. One 256-thread block per 16-row
// block; 8 waves each own a 16-wide column tile; K=128 via 32 x WMMA(K=4).
// Safe in-place (out == A): block reads only its own rows via LDS.
// ---------------------------------------------------------------------------
__global__ __launch_bounds__(256) void gemm_wmma_f32(
    const float* A, const float* __restrict__ W, const float* __restrict__ bias,
    float* out, int n, int apply_leaky)
{
    __shared__ float As[16][DMODEL + 4];   // +4 pad: LDS bank-conflict free

    const int rb = blockIdx.x;
    const long rowBase = (long)rb * 16;
    const int t = threadIdx.x;

    // stage A block (16x128) into LDS, fusing previous layer's LeakyReLU
    for (int i = 0; i < 8; ++i) {
        int idx = t + i * 256;             // 0..2047
        int r = idx >> 7;
        int c = idx & (DMODEL - 1);
        long rr = rowBase + r;
        if (rr >= n) rr = n - 1;           // clamp (keeps EXEC uniform)
        float v = A[rr * DMODEL + c];
        if (apply_leaky) v = leaky(v);
        As[r][c] = v;
    }
    __syncthreads();

    const int wave = t >> 5;               // 0..7 -> column tile
    const int lane = t & 31;
    const int m    = lane & 15;            // row within tile (A) / col (B,C)
    const int half = lane >> 4;            // 0 or 1
    const int ncol = wave * 16 + m;        // output column for this lane

    v8f acc = {0.f, 0.f, 0.f, 0.f, 0.f, 0.f, 0.f, 0.f};

    // A 16x4: lane(m,half): VGPR0=K(2*half), VGPR1=K(2*half+1)
    // B 4x16: lane(n,half): VGPR0=K(2*half), VGPR1=K(2*half+1)
    for (int kb = 0; kb < DMODEL; kb += 4) {
        v2f a, b;
        a.x = As[m][kb + 2 * half];
        a.y = As[m][kb + 2 * half + 1];
        b.x = W[(kb + 2 * half) * DMODEL + ncol];
        b.y = W[(kb + 2 * half + 1) * DMODEL + ncol];
        acc = __builtin_amdgcn_wmma_f32_16x16x4_f32(
            /*neg_a=*/false, a, /*neg_b=*/false, b,
            /*c_mod=*/(short)0, acc, /*reuse_a=*/false, /*reuse_b=*/false);
    }

    const float bv = bias[ncol];
    // C layout: acc[r] -> row (rowBase + r + half*8), col ncol
    for (int r = 0; r < 8; ++r) {
        long orow = rowBase + r + half * 8;
        if (orow < n) out[orow * DMODEL + ncol] = acc[r] + bv;
    }
}

// ---------------------------------------------------------------------------
// Degree / normalization
// ---------------------------------------------------------------------------
__global__ void deg_init(float* deg, int n) {
    int i = blockIdx.x * blockDim.x + threadIdx.x;
    if (i < n) deg[i] = 1.0f;              // self loop
}
__global__ void deg_edges(const int* __restrict__ dst, float* deg, int e) {
    int i = blockIdx.x * blockDim.x + threadIdx.x;
    if (i < e) atomicAdd(&deg[dst[i]], 1.0f);
}
__global__ void deg_rsqrt(float* deg, int n) {
    int i = blockIdx.x * blockDim.x + threadIdx.x;
    if (i < n) deg[i] = rsqrtf(fmaxf(deg[i], 1.0f));   // now holds dinv
}

// agg[n][:] = hlin[n][:] * dinv[n]^2   (self-loop term initializes the sum)
__global__ void agg_selfloop(const float* __restrict__ hlin,
                             const float* __restrict__ dinv,
                             float* __restrict__ agg, int n)
{
    long i = (long)blockIdx.x * blockDim.x + threadIdx.x;   // float4 index
    long total = (long)n * (DMODEL / 4);
    if (i >= total) return;
    long node = i >> 5;                    // 32 float4 per node
    float s = dinv[node]; s *= s;
    float4 v = ((const float4*)hlin)[i];
    v.x *= s; v.y *= s; v.z *= s; v.w *= s;
    ((float4*)agg)[i] = v;
}

// per-edge: agg[dst] += hlin[src] * dinv[src]*dinv[dst]
// 32 lanes per edge, float4 gather per lane (coalesced 512B/edge), f32 atomics
__global__ __launch_bounds__(256) void scatter_edges(
    const float* __restrict__ hlin, const int* __restrict__ src,
    const int* __restrict__ dst, const float* __restrict__ dinv,
    float* __restrict__ agg, int e)
{
    long gid = (long)blockIdx.x * 256 + threadIdx.x;
    long edge = gid >> 5;
    int lane = (int)(gid & 31);
    if (edge >= e) return;
    int s = src[edge], d = dst[edge];
    float w = dinv[s] * dinv[d];
    float4 v = ((const float4*)(hlin + (long)s * DMODEL))[lane];
    float* o = agg + (long)d * DMODEL + lane * 4;
    atomicAdd(o + 0, v.x * w);
    atomicAdd(o + 1, v.y * w);
    atomicAdd(o + 2, v.z * w);
    atomicAdd(o + 3, v.w * w);
}

// ---------------------------------------------------------------------------
// Pooling
// ---------------------------------------------------------------------------
__global__ void pool_init(unsigned* hkey, float* hsum, float* counts) {
    int i = blockIdx.x * blockDim.x + threadIdx.x;
    if (i < NGRAPH * DMODEL) { hkey[i] = fenc(-__builtin_inff()); hsum[i] = 0.f; }
    if (i < NGRAPH) counts[i] = 0.f;
}

__global__ void pool_nodes(const float* __restrict__ h, const int* __restrict__ batch,
                           unsigned* __restrict__ hkey, float* __restrict__ hsum,
                           float* __restrict__ counts, int n)
{
    long gid = (long)blockIdx.x * blockDim.x + threadIdx.x;
    long node = gid >> 7;
    int f = (int)(gid & (DMODEL - 1));
    if (node >= n) return;
    int g = batch[node];
    float v = leaky(h[node * DMODEL + f]);     // final layer's LeakyReLU fused here
    atomicMax(&hkey[g * DMODEL + f], fenc(v));
    atomicAdd(&hsum[g * DMODEL + f], v);
    if (f == 0) atomicAdd(&counts[g], 1.0f);
}

// out[g][c] = blin[c] + sum_k max_k*Wlin[k][c] + sum_k mean_k*Wlin[128+k][c]
__global__ __launch_bounds__(256) void head_kernel(
    const unsigned* __restrict__ hkey, const float* __restrict__ hsum,
    const float* __restrict__ counts, const float* __restrict__ Wlin,
    const float* __restrict__ blin, float* __restrict__ out)
{
    int t = threadIdx.x;
    int g = blockIdx.x * 16 + (t >> 4);
    int c = t & 15;
    if (g >= NGRAPH) return;
    float inv = 1.0f / fmaxf(counts[g], 1.0f);
    float acc = blin[c];
    for (int k = 0; k < DMODEL; ++k)
        acc += fdec(hkey[g * DMODEL + k]) * Wlin[k * NCLASS + c];
    for (int k = 0; k < DMODEL; ++k)
        acc += (hsum[g * DMODEL + k] * inv) * Wlin[(DMODEL + k) * NCLASS + c];
    out[g * NCLASS + c] = acc;
}

// ---------------------------------------------------------------------------
// Host-side orchestration
// ---------------------------------------------------------------------------
static inline long cdiv(long a, long b) { return (a + b - 1) / b; }

extern "C" void kernel_launch(void* const* d_in, const int* in_sizes, int n_in,
                              void* d_out, int out_size, void* d_ws, size_t ws_size,
                              hipStream_t stream)
{
    const float* x     = (const float*)d_in[0];
    const int*   ei    = (const int*)d_in[1];       // [2, E] int32
    const int*   batch = (const int*)d_in[2];
    const float* Wl[5], *bl[5];
    for (int i = 0; i < 5; ++i) {
        Wl[i] = (const float*)d_in[3 + 2 * i];
        bl[i] = (const float*)d_in[4 + 2 * i];
    }
    const float* Wlin = (const float*)d_in[13];
    const float* blin = (const float*)d_in[14];

    const int N = in_sizes[2];             // 100000
    const int E = in_sizes[1] / 2;         // 3200000
    const int* src = ei;
    const int* dst = ei + E;

    // workspace layout
    char* ws = (char*)d_ws;
    size_t off = 0;
    auto take = [&](size_t bytes) { void* p = ws + off; off += (bytes + 255) & ~(size_t)255; return p; };
    float*    bufA   = (float*)take((size_t)N * DMODEL * sizeof(float));
    float*    bufB   = (float*)take((size_t)N * DMODEL * sizeof(float));
    float*    dinv   = (float*)take((size_t)N * sizeof(float));
    unsigned* hkey   = (unsigned*)take((size_t)NGRAPH * DMODEL * sizeof(unsigned));
    float*    hsum   = (float*)take((size_t)NGRAPH * DMODEL * sizeof(float));
    float*    counts = (float*)take((size_t)NGRAPH * sizeof(float));
    (void)ws_size;

    const int rowBlocks = (int)cdiv(N, 16);
    const long nElem4 = (long)N * (DMODEL / 4);

    // --- normalization: dinv = rsqrt(1 + indegree) ---
    deg_init  <<<cdiv(N, 256), 256, 0, stream>>>(dinv, N);
    deg_edges <<<cdiv(E, 256), 256, 0, stream>>>(dst, dinv, E);
    deg_rsqrt <<<cdiv(N, 256), 256, 0, stream>>>(dinv, N);

    // --- 5 GCN layers (ping-pong bufA/bufB; GEMM in-place after layer 0) ---
    // layer 0: x -> bufA (lin), bufA -> bufB (agg)
    gemm_wmma_f32 <<<rowBlocks, 256, 0, stream>>>(x, Wl[0], bl[0], bufA, N, 0);
    agg_selfloop  <<<cdiv(nElem4, 256), 256, 0, stream>>>(bufA, dinv, bufB, N);
    scatter_edges <<<cdiv((long)E * 32, 256), 256, 0, stream>>>(bufA, src, dst, dinv, bufB, E);

    float* h = bufB;                        // current aggregated features
    float* other = bufA;
    for (int i = 1; i < 5; ++i) {
        // lin: leaky(h) @ Wi + bi, in-place into h (safe per 16-row block)
        gemm_wmma_f32 <<<rowBlocks, 256, 0, stream>>>(h, Wl[i], bl[i], h, N, 1);
        // agg: other = selfloop + scatter of h
        agg_selfloop  <<<cdiv(nElem4, 256), 256, 0, stream>>>(h, dinv, other, N);
        scatter_edges <<<cdiv((long)E * 32, 256), 256, 0, stream>>>(h, src, dst, dinv, other, E);
        float* tmp = h; h = other; other = tmp;
    }

    // --- pooling (LeakyReLU of last layer fused into pool read) ---
    pool_init  <<<cdiv(NGRAPH * DMODEL, 256), 256, 0, stream>>>(hkey, hsum, counts);
    pool_nodes <<<cdiv((long)N * DMODEL, 256), 256, 0, stream>>>(h, batch, hkey, hsum, counts, N);

    // --- head ---
    head_kernel <<<NGRAPH / 16, 256, 0, stream>>>(hkey, hsum, counts, Wlin, blin, (float*)d_out);
    (void)n_in; (void)out_size;
}